// VAE_48215302865303
// MI455X (gfx1250) — compile-verified
//
#include <hip/hip_runtime.h>

typedef __attribute__((ext_vector_type(2))) float v2f;
typedef __attribute__((ext_vector_type(4))) float v4f;
typedef __attribute__((ext_vector_type(8))) float v8f;
typedef __attribute__((ext_vector_type(4))) int v4i;

__device__ __forceinline__ v8f wmma_f32(v2f a, v2f b, v8f c) {
  // (neg_a, A, neg_b, B, c_mod, C, reuse_a, reuse_b)
  return __builtin_amdgcn_wmma_f32_16x16x4_f32(false, a, false, b, (short)0, c,
                                               false, false);
}

#if __has_builtin(__builtin_amdgcn_global_load_async_to_lds_b128)
#define HAS_ASYNC_LDS 1
// DMA 16 bytes per lane: global -> LDS, tracked by ASYNCcnt (no VGPR data).
// Signature (from clang diagnostic): param0 = AS1 v4i* (global src).
__device__ __forceinline__ void async_cp16(const float* g, float* l) {
  __builtin_amdgcn_global_load_async_to_lds_b128(
      (__attribute__((address_space(1))) v4i*)g,
      (__attribute__((address_space(3))) v4i*)l, 0, 0);
}
#else
#define HAS_ASYNC_LDS 0
#endif

__device__ __forceinline__ void wait_async_all() {
#if HAS_ASYNC_LDS
  asm volatile("s_wait_asynccnt 0x0" ::: "memory");
#endif
}

#define MBLK 64
#define NBLK 128
#define KBLK 32
#define AS_STRIDE 36   // 36*4 = 144 B: 16B-aligned rows, even for b64 pairs
#define BS_STRIDE 132  // 132*4 = 528 B: 16B-aligned rows

// C[M,N] = act(A[M,K] @ W[K,N] + bias), row-major.
// Block: 256 threads / 8 waves -> 64x128 tile; wave -> 32x32 (4 x v8f accs).
// Double-buffered LDS; tiles staged by async Global->LDS DMA (ASYNCcnt).
// Requires M % 64 == 0, K % 4 == 0, N % 4 == 0.
template <int RELU>
__global__ __launch_bounds__(256, 2) void gemm_bias_kernel(
    const float* __restrict__ A, const float* __restrict__ W,
    const float* __restrict__ bias, float* __restrict__ C, int M, int K,
    int N) {
  __shared__ float As[2][MBLK * AS_STRIDE];
  __shared__ float Bs[2][KBLK * BS_STRIDE];

  const int tid = (int)threadIdx.x;
  const int lane = tid & 31;
  const int waveId = tid >> 5;
  const int rowBase = (int)blockIdx.y * MBLK;
  const int colBase = (int)blockIdx.x * NBLK;

  // Staging geometry (all 256 threads):
  const int aRow = tid >> 3;         // 0..31 (and +32)
  const int aK4 = (tid & 7) << 2;    // 0,4,..,28
  const int bCol = (tid & 31) << 2;  // 0,4,..,124
  const int bRow = tid >> 5;         // 0..7 (and +8,+16,+24)
  const int gcol = colBase + bCol;
  const bool bColOk = gcol < N;  // N % 4 == 0 -> whole float4 in/out

  // Compute geometry (per wave):
  const int wr = (waveId & 1) << 5;   // 0 / 32
  const int wc = (waveId >> 1) << 5;  // 0 / 32 / 64 / 96
  const int m = lane & 15;
  const int koff = (lane >> 4) << 1;  // lanes 16..31 hold K pair +2,+3

  v8f acc00 = {}, acc01 = {}, acc10 = {}, acc11 = {};

  // Stage K-block kb into LDS buffer buf (zero-fill ragged K / N edges).
  auto stage = [&](int kb, int buf) {
    const int k0 = kb * KBLK;
    float* as = As[buf];
    float* bs = Bs[buf];
#if HAS_ASYNC_LDS
    if (k0 + aK4 < K) {  // K % 4 == 0 -> whole float4 valid
      async_cp16(A + (size_t)(rowBase + aRow) * K + (k0 + aK4),
                 as + aRow * AS_STRIDE + aK4);
      async_cp16(A + (size_t)(rowBase + aRow + 32) * K + (k0 + aK4),
                 as + (aRow + 32) * AS_STRIDE + aK4);
    } else {
      *(v4f*)(as + aRow * AS_STRIDE + aK4) = (v4f){};
      *(v4f*)(as + (aRow + 32) * AS_STRIDE + aK4) = (v4f){};
    }
    const float* Bg = W + (size_t)(k0 + bRow) * N + gcol;
#pragma unroll
    for (int r = 0; r < KBLK; r += 8) {
      if (bColOk && (k0 + bRow + r) < K)
        async_cp16(Bg + (size_t)r * N, bs + (bRow + r) * BS_STRIDE + bCol);
      else
        *(v4f*)(bs + (bRow + r) * BS_STRIDE + bCol) = (v4f){};
    }
#else
    v4f a0 = {}, a1 = {};
    if (k0 + aK4 < K) {
      a0 = *(const v4f*)(A + (size_t)(rowBase + aRow) * K + (k0 + aK4));
      a1 = *(const v4f*)(A + (size_t)(rowBase + aRow + 32) * K + (k0 + aK4));
    }
    v4f b0 = {}, b1 = {}, b2 = {}, b3 = {};
    if (bColOk) {
      const float* Bg = W + (size_t)(k0 + bRow) * N + gcol;
      if (k0 + bRow < K) b0 = *(const v4f*)(Bg);
      if (k0 + bRow + 8 < K) b1 = *(const v4f*)(Bg + (size_t)8 * N);
      if (k0 + bRow + 16 < K) b2 = *(const v4f*)(Bg + (size_t)16 * N);
      if (k0 + bRow + 24 < K) b3 = *(const v4f*)(Bg + (size_t)24 * N);
    }
    *(v4f*)(as + aRow * AS_STRIDE + aK4) = a0;
    *(v4f*)(as + (aRow + 32) * AS_STRIDE + aK4) = a1;
    *(v4f*)(bs + bRow * BS_STRIDE + bCol) = b0;
    *(v4f*)(bs + (bRow + 8) * BS_STRIDE + bCol) = b1;
    *(v4f*)(bs + (bRow + 16) * BS_STRIDE + bCol) = b2;
    *(v4f*)(bs + (bRow + 24) * BS_STRIDE + bCol) = b3;
#endif
  };

  // Branch-free WMMA chain over one staged K-block.
  auto compute = [&](int buf) {
    const float* as = As[buf];
    const float* bs = Bs[buf];
#pragma unroll
    for (int kk = 0; kk < KBLK; kk += 4) {
      const int ka = kk + koff;
      v2f fa0 = *(const v2f*)(as + (wr + m) * AS_STRIDE + ka);
      v2f fa1 = *(const v2f*)(as + (wr + 16 + m) * AS_STRIDE + ka);
      v2f fb0, fb1;
      fb0.x = bs[ka * BS_STRIDE + wc + m];
      fb0.y = bs[(ka + 1) * BS_STRIDE + wc + m];
      fb1.x = bs[ka * BS_STRIDE + wc + 16 + m];
      fb1.y = bs[(ka + 1) * BS_STRIDE + wc + 16 + m];
      acc00 = wmma_f32(fa0, fb0, acc00);
      acc01 = wmma_f32(fa0, fb1, acc01);
      acc10 = wmma_f32(fa1, fb0, acc10);
      acc11 = wmma_f32(fa1, fb1, acc11);
    }
  };

  const int numK = (K + KBLK - 1) / KBLK;
  stage(0, 0);
  wait_async_all();
  __syncthreads();
  for (int kb = 0; kb < numK; ++kb) {
    const int cur = kb & 1;
    if (kb + 1 < numK) stage(kb + 1, cur ^ 1);  // DMA overlaps compute
    compute(cur);
    wait_async_all();  // own DMA into next buffer done
    __syncthreads();   // everyone's DMA + zero-fill visible
  }

  // ---- epilogue: bias + activation + store (divergence OK here) ----
  const int crowOff = (lane >> 4) << 3;  // 0 / 8
  auto store_tile = [&](v8f acc, int tr, int tc) {
    const int col = colBase + wc + tc * 16 + m;
    if (col < N) {
      const float bv = bias[col];
      const int r = rowBase + wr + tr * 16 + crowOff;
#pragma unroll
      for (int v = 0; v < 8; ++v) {
        float val = acc[v] + bv;
        if (RELU) val = fmaxf(val, 0.0f);
        C[(size_t)(r + v) * N + col] = val;
      }
    }
  };
  store_tile(acc00, 0, 0);
  store_tile(acc01, 0, 1);
  store_tile(acc10, 1, 0);
  store_tile(acc11, 1, 1);
}

#define NN 68
#define KK 32
#define DD (NN * NN)

// Per-batch: mu = h11@fc21+b, logvar = h12@fc22+b, z = mu+eps*exp(.5lv),
// x_nodes = z@z2n+b.  One block (128 threads) per batch element.
__global__ __launch_bounds__(128) void heads_kernel(
    const float* __restrict__ h11, const float* __restrict__ h12,
    const float* __restrict__ fc21_W, const float* __restrict__ fc21_b,
    const float* __restrict__ fc22_W, const float* __restrict__ fc22_b,
    const float* __restrict__ z2n_W, const float* __restrict__ z2n_b,
    const float* __restrict__ eps, float* __restrict__ mu_out,
    float* __restrict__ logvar_out, float* __restrict__ x_nodes) {
  const int b = blockIdx.x;
  const int t = threadIdx.x;
  __shared__ float s11[128], s12[128], sz[NN];
  s11[t] = h11[(size_t)b * 128 + t];
  s12[t] = h12[(size_t)b * 128 + t];
  __syncthreads();
  if (t < NN) {
    float m = fc21_b[t], lv = fc22_b[t];
    for (int j = 0; j < 128; ++j) {
      m += s11[j] * fc21_W[j * NN + t];
      lv += s12[j] * fc22_W[j * NN + t];
    }
    mu_out[(size_t)b * NN + t] = m;
    logvar_out[(size_t)b * NN + t] = lv;
    sz[t] = m + eps[(size_t)b * NN + t] * expf(0.5f * lv);
  }
  __syncthreads();
  if (t < NN) {
    float acc = z2n_b[t];
    for (int j = 0; j < NN; ++j) acc += sz[j] * z2n_W[j * NN + t];
    x_nodes[(size_t)b * NN + t] = acc;
  }
}

// Three GCN gather layers + outer product x3 x3^T -> xf2[b, 68*68].
__global__ __launch_bounds__(128) void gcn_outer_kernel(
    const float* __restrict__ x_nodes, const int* __restrict__ adj,
    const float* __restrict__ W1, const float* __restrict__ b1,
    const float* __restrict__ W2, const float* __restrict__ b2,
    const float* __restrict__ W3, const float* __restrict__ b3,
    float* __restrict__ xf2) {
  const int b = blockIdx.x;
  const int t = threadIdx.x;
  __shared__ float cur[NN];
  if (t < NN) cur[t] = x_nodes[(size_t)b * NN + t];
  __syncthreads();
  const float* Ws[3] = {W1, W2, W3};
  const float* bs[3] = {b1, b2, b3};
  for (int l = 0; l < 3; ++l) {
    float v = 0.0f;
    if (t < NN) {
      float acc = bs[l][t];
      for (int k = 0; k < KK; ++k)
        acc += cur[adj[t * KK + k]] * Ws[l][t * KK + k];
      v = 1.0f / (1.0f + expf(-acc));
    }
    __syncthreads();
    if (t < NN) cur[t] = v;
    __syncthreads();
  }
  float* out = xf2 + (size_t)b * DD;
  for (int idx = t; idx < DD; idx += blockDim.x)
    out[idx] = cur[idx / NN] * cur[idx % NN];
}

extern "C" void kernel_launch(void* const* d_in, const int* in_sizes, int n_in,
                              void* d_out, int out_size, void* d_ws,
                              size_t ws_size, hipStream_t stream) {
  const float* x = (const float*)d_in[0];
  const float* eps = (const float*)d_in[1];
  const int* adj = (const int*)d_in[2];
  const float* fc11_W = (const float*)d_in[3];
  const float* fc11_b = (const float*)d_in[4];
  const float* fc12_W = (const float*)d_in[5];
  const float* fc12_b = (const float*)d_in[6];
  const float* fc111_W = (const float*)d_in[7];
  const float* fc111_b = (const float*)d_in[8];
  const float* fc222_W = (const float*)d_in[9];
  const float* fc222_b = (const float*)d_in[10];
  const float* fc21_W = (const float*)d_in[11];
  const float* fc21_b = (const float*)d_in[12];
  const float* fc22_W = (const float*)d_in[13];
  const float* fc22_b = (const float*)d_in[14];
  const float* z2n_W = (const float*)d_in[15];
  const float* z2n_b = (const float*)d_in[16];
  const float* W1 = (const float*)d_in[17];
  const float* b1 = (const float*)d_in[18];
  const float* W2 = (const float*)d_in[19];
  const float* b2 = (const float*)d_in[20];
  const float* W3 = (const float*)d_in[21];
  const float* b3 = (const float*)d_in[22];
  const float* out_W = (const float*)d_in[23];
  const float* out_b = (const float*)d_in[24];

  const int B = 4096;
  float* recon = (float*)d_out;         // B*D
  float* mu = recon + (size_t)B * DD;   // B*N
  float* logvar = mu + (size_t)B * NN;  // B*N

  float* ws = (float*)d_ws;
  float* h11 = ws;                         // B*1024 (dead before xf2 written)
  float* h12 = ws + (size_t)B * 1024;      // B*1024 (dead before xf2 written)
  float* xf2 = ws;                         // B*D, overlaps h11/h12
  float* h111 = ws + (size_t)B * DD;       // B*128
  float* h222 = h111 + (size_t)B * 128;    // B*128
  float* xnodes = h222 + (size_t)B * 128;  // B*N

  const dim3 blk(256);
  const int mBlks = B / MBLK;  // 64
  // fc11: 4096x4624 @ 4624x1024, relu
  gemm_bias_kernel<1><<<dim3(1024 / NBLK, mBlks), blk, 0, stream>>>(
      x, fc11_W, fc11_b, h11, B, DD, 1024);
  // fc111: 4096x1024 @ 1024x128, relu
  gemm_bias_kernel<1><<<dim3(1, mBlks), blk, 0, stream>>>(
      h11, fc111_W, fc111_b, h111, B, 1024, 128);
  // fc12: 4096x4624 @ 4624x1024, relu
  gemm_bias_kernel<1><<<dim3(1024 / NBLK, mBlks), blk, 0, stream>>>(
      x, fc12_W, fc12_b, h12, B, DD, 1024);
  // fc222: 4096x1024 @ 1024x128, relu
  gemm_bias_kernel<1><<<dim3(1, mBlks), blk, 0, stream>>>(
      h12, fc222_W, fc222_b, h222, B, 1024, 128);
  // heads: mu/logvar/z/x_nodes
  heads_kernel<<<dim3(B), dim3(128), 0, stream>>>(
      h111, h222, fc21_W, fc21_b, fc22_W, fc22_b, z2n_W, z2n_b, eps, mu,
      logvar, xnodes);
  // 3 GCN layers + outer product
  gcn_outer_kernel<<<dim3(B), dim3(128), 0, stream>>>(xnodes, adj, W1, b1, W2,
                                                      b2, W3, b3, xf2);
  // recon: 4096x4624 @ 4624x4624 (N tail 16 cols handled in-kernel)
  gemm_bias_kernel<0><<<dim3((DD + NBLK - 1) / NBLK, mBlks), blk, 0, stream>>>(
      xf2, out_W, out_b, recon, B, DD, DD);
}